// SparseLinear_34686155883133
// MI455X (gfx1250) — compile-verified
//
#include <hip/hip_runtime.h>

typedef __attribute__((ext_vector_type(2))) float v2f;
typedef __attribute__((ext_vector_type(8))) float v8f;

#define OUTF 128  // fast path specialized for OUT_F == 128

// ---------------- Kernel 0: tiled transpose W[out_f][in_f] -> Wt[in_f][out_f]
__global__ void k_transpose(const float* __restrict__ w, float* __restrict__ wt,
                            int in_f, int out_f) {
    __shared__ float tile[32][33];
    int bx = blockIdx.x * 32;            // along in_f
    int by = blockIdx.y * 32;            // along out_f
    int tx = threadIdx.x, ty = threadIdx.y;  // 32 x 8
    for (int i = ty; i < 32; i += 8) {
        int r = by + i, c = bx + tx;
        tile[i][tx] = (r < out_f && c < in_f) ? w[(size_t)r * in_f + c] : 0.f;
    }
    __syncthreads();
    for (int i = ty; i < 32; i += 8) {
        int r = bx + i, c = by + tx;     // Wt[r][c] = W[c][r]
        if (r < in_f && c < out_f) wt[(size_t)r * out_f + c] = tile[tx][i];
    }
}

// ---------------- Kernel 1: zero an int region
__global__ void k_zero(int* __restrict__ p, long long n) {
    long long i = (long long)blockIdx.x * blockDim.x + threadIdx.x;
    if (i < n) p[i] = 0;
}

// ---------------- Kernel 2: histogram of rows
__global__ void k_hist(const int* __restrict__ rows, int* __restrict__ counts, int nnz) {
    int i = blockIdx.x * blockDim.x + threadIdx.x;
    if (i < nnz) atomicAdd(&counts[rows[i]], 1);
}

// ---------------- Kernel 3: group-granular exclusive scan (single 1024-thread block)
// Each 16-row group's packed segment is padded to a multiple of 4 entries so the
// accumulate kernel needs no tail guards. gs[g] = padded start of group g.
__global__ void k_scan(const int* __restrict__ counts, int* __restrict__ cursor,
                       int* __restrict__ gs, int n_rows, int ngroups) {
    __shared__ int part[1024];
    int t = threadIdx.x;
    int chunk = (ngroups + 1023) / 1024;
    int gbeg = t * chunk;
    int gend = min(gbeg + chunk, ngroups);
    int s = 0;
    for (int g = gbeg; g < gend; ++g) {
        int r0 = g * 16, r1 = min(r0 + 16, n_rows);
        int gsz = 0;
        for (int r = r0; r < r1; ++r) gsz += counts[r];
        s += (gsz + 3) & ~3;
    }
    part[t] = s;
    __syncthreads();
    for (int off = 1; off < 1024; off <<= 1) {
        int v = part[t];
        int add = (t >= off) ? part[t - off] : 0;
        __syncthreads();
        part[t] = v + add;
        __syncthreads();
    }
    int run = (t == 0) ? 0 : part[t - 1];
    for (int g = gbeg; g < gend; ++g) {
        gs[g] = run;
        int r0 = g * 16, r1 = min(r0 + 16, n_rows);
        int run2 = run, gsz = 0;
        for (int r = r0; r < r1; ++r) {
            cursor[r] = run2;
            run2 += counts[r];
            gsz += counts[r];
        }
        run += (gsz + 3) & ~3;
    }
    if (t == 1023) gs[ngroups] = run;   // `run` = padded total, after adding t=1023's own groups
}

// ---------------- Kernel 4: bin nonzeros by row into interleaved (owner<<16|col, val) records
__global__ void k_bin(const int* __restrict__ rows, const int* __restrict__ cols,
                      const float* __restrict__ vals, int* __restrict__ cursor,
                      uint2* __restrict__ pk, int nnz) {
    int i = blockIdx.x * blockDim.x + threadIdx.x;
    if (i >= nnz) return;
    int r = rows[i];
    int p = atomicAdd(&cursor[r], 1);
    uint2 rec;
    rec.x = ((unsigned)(r & 15) << 16) | (unsigned)cols[i];
    rec.y = __float_as_uint(vals[i]);
    pk[p] = rec;
}

// ---------------- Kernel 5: one wave per 16-row group; guard-free rank-4 WMMA updates
__global__ void k_spmm_wmma(const uint2* __restrict__ pk, const int* __restrict__ gs,
                            const float* __restrict__ wt, const float* __restrict__ bias,
                            float* __restrict__ out, int n_rows) {
    int wave = (int)((blockIdx.x * blockDim.x + threadIdx.x) >> 5);
    int lane = threadIdx.x & 31;
    int ngroups = (n_rows + 15) >> 4;
    if (wave >= ngroups) return;

    int r0   = wave << 4;
    int p    = gs[wave];
    int pend = gs[wave + 1];   // padded: (pend - p) % 4 == 0, pad records are val=0

    int half = lane >> 4;      // 0: lanes 0-15 handle K=0,1 ; 1: lanes 16-31 handle K=2,3
    int m    = lane & 15;      // local M/N index per ISA striping
    int koff = half << 1;      // even record index -> 16B-aligned b128 load

    v8f acc[8] = {};           // 8 N-tiles of 16 -> 128 output columns

    if (p < pend) {
        const uint4* lanep = (const uint4*)(pk + p + koff);   // this lane's 2 records
        uint4 e = *lanep;                                     // chunk p
        for (; p < pend; p += 4) {
            uint4 e_next = lanep[2];   // chunk p+4 (tail chunk zero-padded; safe overread)
            lanep += 2;

            int c0 = (int)(e.x & 0xFFFFu), o0 = (int)(e.x >> 16);
            int c1 = (int)(e.z & 0xFFFFu), o1 = (int)(e.z >> 16);
            float v0 = __uint_as_float(e.y);
            float v1 = __uint_as_float(e.w);

            // A[m][k] = v_k iff nonzero k belongs to local row m (pad: v=0)
            v2f a;
            a.x = (o0 == m) ? v0 : 0.f;
            a.y = (o1 == m) ? v1 : 0.f;

            const float* wr0 = wt + (size_t)c0 * OUTF + m;   // B[k][n] = Wt[c_k][n]
            const float* wr1 = wt + (size_t)c1 * OUTF + m;
#pragma unroll
            for (int t = 0; t < 8; ++t) {
                v2f b;
                b.x = wr0[t * 16];
                b.y = wr1[t * 16];
                acc[t] = __builtin_amdgcn_wmma_f32_16x16x4_f32(
                    false, a, false, b, (short)0, acc[t], false, false);
            }
            e = e_next;
        }
    }

    // Writeout: C/D layout -> lane<16: (M=j, N=m), lane>=16: (M=j+8, N=m)
    float bv[8];
#pragma unroll
    for (int t = 0; t < 8; ++t) bv[t] = bias[t * 16 + m];

    if (r0 + 16 <= n_rows) {
        // full group: guard-free straight-line stores (immediate offsets t*64B)
        float* basep = out + (size_t)(r0 + half * 8) * OUTF + m;
#pragma unroll
        for (int j = 0; j < 8; ++j) {
            float* rp = basep + (size_t)j * OUTF;
#pragma unroll
            for (int t = 0; t < 8; ++t) {
                rp[t * 16] = acc[t][j] + bv[t];
            }
        }
    } else {
        // tail group: one guard per row (not per store)
#pragma unroll
        for (int j = 0; j < 8; ++j) {
            int row = r0 + j + half * 8;
            if (row < n_rows) {
                float* rp = out + (size_t)row * OUTF + m;
#pragma unroll
                for (int t = 0; t < 8; ++t) {
                    rp[t * 16] = acc[t][j] + bv[t];
                }
            }
        }
    }
}

// ---------------- Fallback (workspace too small): bias init + atomic scatter
__global__ void k_bias_init(float* __restrict__ out, const float* __restrict__ bias,
                            int total, int out_f) {
    int i = blockIdx.x * blockDim.x + threadIdx.x;
    if (i < total) out[i] = bias[i % out_f];
}

__global__ void k_atomic_scatter(const int* __restrict__ rows, const int* __restrict__ cols,
                                 const float* __restrict__ vals, const float* __restrict__ w,
                                 float* __restrict__ out, int nnz, int in_f, int out_f) {
    long long tid = (long long)blockIdx.x * blockDim.x + threadIdx.x;
    long long total = (long long)nnz * out_f;
    if (tid >= total) return;
    int i = (int)(tid / out_f);
    int o = (int)(tid % out_f);
    int r = rows[i], c = cols[i];
    atomicAdd(&out[(size_t)r * out_f + o], vals[i] * w[(size_t)o * in_f + c]);
}

static inline size_t align16(size_t x) { return (x + 15) & ~(size_t)15; }

extern "C" void kernel_launch(void* const* d_in, const int* in_sizes, int n_in,
                              void* d_out, int out_size, void* d_ws, size_t ws_size,
                              hipStream_t stream) {
    const int*   rows = (const int*)d_in[0];
    const int*   cols = (const int*)d_in[1];
    const float* vals = (const float*)d_in[2];
    const float* wgt  = (const float*)d_in[3];
    const float* bias = (const float*)d_in[4];

    const int nnz    = in_sizes[0];
    const int out_f  = in_sizes[4];
    const int in_f   = in_sizes[3] / out_f;
    const int n_rows = out_size / out_f;
    float* out = (float*)d_out;

    const int ngroups = (n_rows + 15) / 16;
    // capacity: nnz + up to 3 pad per group + 1 extra tail chunk for pipelined overread
    const long long cap = (long long)nnz + 4LL * ngroups + 4;

    size_t wt_b     = align16((size_t)in_f * out_f * 4);
    size_t counts_b = align16((size_t)n_rows * 4);
    size_t cursor_b = align16((size_t)n_rows * 4);
    size_t gs_b     = align16(((size_t)ngroups + 1) * 4);
    size_t pk_b     = align16((size_t)cap * 8);
    size_t need = wt_b + counts_b + cursor_b + gs_b + pk_b;

    if (out_f == OUTF && ws_size >= need) {
        char* ws = (char*)d_ws;
        size_t off = 0;
        float* wt     = (float*)(ws + off);  off += wt_b;
        int*   counts = (int*)(ws + off);    off += counts_b;
        int*   cursor = (int*)(ws + off);    off += cursor_b;
        int*   gs     = (int*)(ws + off);    off += gs_b;
        uint2* pk     = (uint2*)(ws + off);

        dim3 tb(32, 8);
        dim3 tg((in_f + 31) / 32, (out_f + 31) / 32);
        k_transpose<<<tg, tb, 0, stream>>>(wgt, wt, in_f, out_f);

        k_zero<<<(n_rows + 255) / 256, 256, 0, stream>>>(counts, n_rows);
        long long pk_ints = cap * 2;
        k_zero<<<(int)((pk_ints + 255) / 256), 256, 0, stream>>>((int*)pk, pk_ints);

        k_hist<<<(nnz + 255) / 256, 256, 0, stream>>>(rows, counts, nnz);
        k_scan<<<1, 1024, 0, stream>>>(counts, cursor, gs, n_rows, ngroups);
        k_bin<<<(nnz + 255) / 256, 256, 0, stream>>>(rows, cols, vals, cursor, pk, nnz);

        long long threads = (long long)ngroups * 32;   // one wave per group
        int blocks = (int)((threads + 255) / 256);
        k_spmm_wmma<<<blocks, 256, 0, stream>>>(pk, gs, wt, bias, out, n_rows);
    } else {
        k_bias_init<<<(out_size + 255) / 256, 256, 0, stream>>>(out, bias, out_size, out_f);
        long long total = (long long)nnz * out_f;
        int blocks = (int)((total + 255) / 256);
        k_atomic_scatter<<<blocks, 256, 0, stream>>>(rows, cols, vals, wgt, out, nnz, in_f, out_f);
    }
}